// SineGordonIntegrator_66151086293499
// MI455X (gfx1250) — compile-verified
//
#include <hip/hip_runtime.h>
#include <hip/hip_bf16.h>

// Sine-Gordon 2D RK4 integrator for MI455X (gfx1250).
// Laplacian per 16x16 tile computed on the matrix pipe:
//   Lap*dx^2 = T*X + X*T + halo corrections,  T = tridiag(1,-2,1) (symmetric)
// via 8x V_WMMA_F32_16X16X4_F32 per tile (accumulator chained through C).
// Whole working set (~34MB) is L2-resident (192MB L2); DRAM sees only the
// initial load + 84MB of snapshots -> L2-streaming bound, WMMA co-executes
// with v_sin_f32 / AXPY VALU work.

#define NND 1026                 // grid incl. ghost ring
#define NPTS (NND * NND)
#define RING_CELLS 4100          // 2*1026 + 2*1024
#define TILE_BLOCKS 512          // 4096 tiles / 8 waves per block
#define RING_BLOCKS 17           // ceil(4100/256)

typedef __attribute__((ext_vector_type(2))) float v2f;
typedef __attribute__((ext_vector_type(8))) float v8f;

__device__ __forceinline__ float tmat(int i, int j) {
    int d = i - j;
    if (d == 0) return -2.0f;
    if (d == 1 || d == -1) return 1.0f;
    return 0.0f;
}

// D = A(16x4 f32) * B(4x16 f32) + C, wave32. 8-arg form per probe doc.
__device__ __forceinline__ v8f wmma4(v2f a, v2f b, v8f c) {
    return __builtin_amdgcn_wmma_f32_16x16x4_f32(
        /*neg_a=*/false, a, /*neg_b=*/false, b,
        /*c_mod=*/(short)0, c, /*reuse_a=*/false, /*reuse_b=*/false);
}

__device__ __forceinline__ void ring_map(int c, int& i, int& j) {
    if (c < 2 * NND) { i = (c < NND) ? 0 : (NND - 1); j = c % NND; }
    else { int c2 = c - 2 * NND; i = 1 + (c2 & 1023); j = (c2 < 1024) ? 0 : (NND - 1); }
}

// One RK4 stage:  ku = pv ; kv = c2*Lap(pu) - sin(pu)   (Lap==0 on ring)
//   n{u,v} (+)= wdt6 * k     (FIRST initializes from base u,v)
//   q{u,v}  =  base + adt*k  (next stage input; skipped for stage 4)
template <bool FIRST, bool WRITEQ>
__global__ void __launch_bounds__(256)
sg_stage(const float* __restrict__ u, const float* __restrict__ v,
         const float* pu, const float* pv,
         float* __restrict__ nu, float* __restrict__ nv,
         float* qu, float* qv,
         float wdt6, float adt) {
    const float inv_dx2 = (float)(1.0 / ((20.0 / 1025.0) * (20.0 / 1025.0)));
    const int tid  = threadIdx.x;
    const int lane = tid & 31;
    const int half = lane >> 4;   // 0: K{0,1}/M 0-7 rows ; 1: K{2,3}/M 8-15 rows
    const int m    = lane & 15;   // M for A, N for B/C/D

    if (blockIdx.x < TILE_BLOCKS) {
        // ---- interior 16x16 tile per wave, Laplacian via WMMA ----
        const int w   = blockIdx.x * 8 + (tid >> 5);
        const int gi0 = 1 + (w >> 6) * 16;   // tile origin (interior)
        const int gj0 = 1 + (w & 63) * 16;

        // T slices: serve as A (M=m, K=k0,k0+1) and, by symmetry+mirrored
        // layout, as B (K rows k0,k0+1 at N=m).
        v2f tA[4];
#pragma unroll
        for (int s = 0; s < 4; ++s) {
            int k0 = 4 * s + 2 * half;
            tA[s].x = tmat(m, k0);
            tA[s].y = tmat(m, k0 + 1);
        }

        const float* X = pu;
        v8f d = {0.f, 0.f, 0.f, 0.f, 0.f, 0.f, 0.f, 0.f};
        // d += T * X   (vertical second difference, interior rows of tile)
#pragma unroll
        for (int s = 0; s < 4; ++s) {
            int k0 = 4 * s + 2 * half;
            v2f b;
            b.x = X[(gi0 + k0) * NND + gj0 + m];
            b.y = X[(gi0 + k0 + 1) * NND + gj0 + m];
            d = wmma4(tA[s], b, d);
        }
        // d += X * T   (horizontal second difference)
#pragma unroll
        for (int s = 0; s < 4; ++s) {
            int k0 = 4 * s + 2 * half;
            const float* rp = X + (gi0 + m) * NND + gj0 + k0;
            v2f a2; a2.x = rp[0]; a2.y = rp[1];
            d = wmma4(a2, tA[s], d);
        }
        // halo corrections (rank-1 updates on tile edges)
        if (half == 0) d[0] += X[(gi0 - 1) * NND + gj0 + m];      // row 0: +X[-1][n]
        else           d[7] += X[(gi0 + 16) * NND + gj0 + m];     // row 15: +X[16][n]
        if (m == 0) {
#pragma unroll
            for (int r = 0; r < 8; ++r) d[r] += X[(gi0 + 8 * half + r) * NND + (gj0 - 1)];
        }
        if (m == 15) {
#pragma unroll
            for (int r = 0; r < 8; ++r) d[r] += X[(gi0 + 8 * half + r) * NND + (gj0 + 16)];
        }

        // per-element RK stage update; D layout: element r -> (gi0+8*half+r, gj0+m)
        const int base = (gi0 + 8 * half) * NND + gj0 + m;
#pragma unroll
        for (int r = 0; r < 8; ++r) {
            const int idx = base + r * NND;
            const float puc = pu[idx];
            const float ku  = pv[idx];
            const float kv  = inv_dx2 * d[r] - __sinf(puc);
            const float bu = u[idx], bv = v[idx];
            const float au = FIRST ? bu : nu[idx];
            const float av = FIRST ? bv : nv[idx];
            nu[idx] = au + wdt6 * ku;
            nv[idx] = av + wdt6 * kv;
            if (WRITEQ) {
                qu[idx] = bu + adt * ku;
                qv[idx] = bv + adt * kv;
            }
        }
    } else {
        // ---- ghost ring: lap == 0 there -> kv = -sin(pu) ----
        const int c = (blockIdx.x - TILE_BLOCKS) * 256 + tid;
        if (c < RING_CELLS) {
            int i, j; ring_map(c, i, j);
            const int idx = i * NND + j;
            const float ku = pv[idx];
            const float kv = -__sinf(pu[idx]);
            const float bu = u[idx], bv = v[idx];
            const float au = FIRST ? bu : nu[idx];
            const float av = FIRST ? bv : nv[idx];
            nu[idx] = au + wdt6 * ku;
            nv[idx] = av + wdt6 * kv;
            if (WRITEQ) {
                qu[idx] = bu + adt * ku;
                qv[idx] = bv + adt * kv;
            }
        }
    }
}

// Neumann BC: ring <- clamped interior neighbor (row-then-col order == clamp)
__global__ void __launch_bounds__(256)
sg_bc(float* __restrict__ a, float* __restrict__ b) {
    const int c = blockIdx.x * 256 + threadIdx.x;
    if (c >= RING_CELLS) return;
    int i, j; ring_map(c, i, j);
    const int ci = min(max(i, 1), NND - 2);
    const int cj = min(max(j, 1), NND - 2);
    a[i * NND + j] = a[ci * NND + cj];
    b[i * NND + j] = b[ci * NND + cj];
}

// Initial state: copy inputs with BC already applied (clamped indexing)
__global__ void __launch_bounds__(256)
sg_init(const float* __restrict__ u0, const float* __restrict__ v0,
        float* __restrict__ u, float* __restrict__ v) {
    const int c = blockIdx.x * 256 + threadIdx.x;
    if (c >= NPTS) return;
    const int i = c / NND, j = c % NND;
    const int ci = min(max(i, 1), NND - 2);
    const int cj = min(max(j, 1), NND - 2);
    u[c] = u0[ci * NND + cj];
    v[c] = v0[ci * NND + cj];
}

extern "C" void kernel_launch(void* const* d_in, const int* in_sizes, int n_in,
                              void* d_out, int out_size, void* d_ws, size_t ws_size,
                              hipStream_t stream) {
    const float* u0 = (const float*)d_in[0];
    const float* v0 = (const float*)d_in[1];
    float* out = (float*)d_out;

    // workspace: 8 * NPTS floats (~34 MB), all L2-resident during the run
    float* ws = (float*)d_ws;
    float* uA = ws + (size_t)0 * NPTS;
    float* vA = ws + (size_t)1 * NPTS;
    float* uB = ws + (size_t)2 * NPTS;
    float* vB = ws + (size_t)3 * NPTS;
    float* pU = ws + (size_t)4 * NPTS;
    float* pV = ws + (size_t)5 * NPTS;
    float* qU = ws + (size_t)6 * NPTS;
    float* qV = ws + (size_t)7 * NPTS;

    const float dt = (float)(1.0 / 100.0);
    const float dt6 = dt / 6.0f;
    const int grid = TILE_BLOCKS + RING_BLOCKS;

    sg_init<<<(NPTS + 255) / 256, 256, 0, stream>>>(u0, v0, uA, vA);

    float* u = uA; float* v = vA;     // current state
    float* nu = uB; float* nv = vB;   // accumulating next state

    for (int step = 0; step < 100; ++step) {
        // k1: p = (u,v), next-stage input -> (pU,pV)
        sg_stage<true,  true ><<<grid, 256, 0, stream>>>(u, v, u,  v,  nu, nv, pU, pV, dt6 * 1.0f, 0.5f * dt);
        // k2
        sg_stage<false, true ><<<grid, 256, 0, stream>>>(u, v, pU, pV, nu, nv, qU, qV, dt6 * 2.0f, 0.5f * dt);
        // k3
        sg_stage<false, true ><<<grid, 256, 0, stream>>>(u, v, qU, qV, nu, nv, pU, pV, dt6 * 2.0f, 1.0f * dt);
        // k4 (no next-stage output)
        sg_stage<false, false><<<grid, 256, 0, stream>>>(u, v, pU, pV, nu, nv, nullptr, nullptr, dt6 * 1.0f, 0.0f);
        sg_bc<<<RING_BLOCKS, 256, 0, stream>>>(nu, nv);

        { float* t = u; u = nu; nu = t; t = v; v = nv; nv = t; }

        if ((step + 1) % 10 == 0) {
            const int s = (step + 1) / 10 - 1;
            hipMemcpyAsync(out + (size_t)s * NPTS,        u, (size_t)NPTS * sizeof(float),
                           hipMemcpyDeviceToDevice, stream);
            hipMemcpyAsync(out + (size_t)(10 + s) * NPTS, v, (size_t)NPTS * sizeof(float),
                           hipMemcpyDeviceToDevice, stream);
        }
    }
}